// GATGCNNet_56040733278476
// MI455X (gfx1250) — compile-verified
//
// CDNA5 / gfx1250 implementation of the GAT+GCN+biLSTM fusion network.
// - All dense layers on V_WMMA_F32_16X16X4_F32 (precision-exact fp32 matrix pipe;
//   residual blocks alone are ~2 TFLOP fp32 with weights L2-resident).
// - GEMM LDS staging is double-buffered with GLOBAL_LOAD_ASYNC_TO_LDS_B32 +
//   s_wait_asynccnt (CDNA5 async path) where tiles are aligned; guarded
//   synchronous fallback otherwise.
// - fxt1 (1.54 GB weights) is a split-K=80 HBM-bound GEMM (~66us floor @ 23.3 TB/s).
// - Graph scatter/softmax via fp32 L2 atomics + order-preserving uint atomicMax.
// - LSTM recurrence: persistent workgroup per direction, h-state in LDS.
// Workspace requirement: ~0.75 GB.

#include <hip/hip_runtime.h>
#include <hip/hip_bf16.h>
#include <math.h>

typedef __attribute__((ext_vector_type(2))) float v2f;
typedef __attribute__((ext_vector_type(8))) float v8f;

#if defined(__has_builtin)
#if __has_builtin(__builtin_amdgcn_wmma_f32_16x16x4_f32)
#define HAVE_WMMA_F32 1
#endif
#if __has_builtin(__builtin_amdgcn_global_load_async_to_lds_b32)
#define HAVE_ASYNC_LDS 1
#endif
#endif
#ifndef HAVE_WMMA_F32
#define HAVE_WMMA_F32 0
#endif
#ifndef HAVE_ASYNC_LDS
#define HAVE_ASYNC_LDS 0
#endif

#if HAVE_ASYNC_LDS
typedef __attribute__((address_space(1))) int as1_int;
typedef __attribute__((address_space(3))) int as3_int;
__device__ __forceinline__ void async_ld32(const void* g, void* l) {
  __builtin_amdgcn_global_load_async_to_lds_b32((as1_int*)g, (as3_int*)l, 0, 0);
}
#endif

enum {
  Nn = 12800, Ne = 25600, E2c = Ne + Nn,   // edges + self loops
  Bb = 32, Ts = 1000,
  F0 = 78, HEADS = 10, F780 = 780, F1560 = 1560, F3120 = 3120
};

// ---------- helpers ----------
__device__ __forceinline__ unsigned fordu(float f) {
  unsigned u = __float_as_uint(f);
  return (u & 0x80000000u) ? ~u : (u | 0x80000000u);
}
__device__ __forceinline__ float unford(unsigned u) {
  u = (u & 0x80000000u) ? (u & 0x7FFFFFFFu) : ~u;
  return __uint_as_float(u);
}
__device__ __forceinline__ void edge_sd(const int* ei, int e2, int& s, int& d) {
  if (e2 < Ne) { s = ei[e2]; d = ei[Ne + e2]; }
  else         { s = d = e2 - Ne; }
}
__device__ __forceinline__ float sigf(float x) { return 1.0f / (1.0f + expf(-x)); }

// ======================================================================
// Generic fp32 WMMA GEMM:  C[M,N] = act( A[M,K] @ W + bias1 + bias2 )
//   WT=false: W is [K,N] row major.  WT=true: W is [N,K] row major (B^T).
//   splitK>1 + atomicOut: partial sums atomically accumulated into C.
// Tiling: 128x64 macro tile, BK=16, 8 waves, each wave a 32x32 sub-tile
// (2x2 fragments of v_wmma_f32_16x16x4_f32). Aligned tiles stream through
// a double-buffered async-to-LDS pipeline (ASYNCcnt); ragged tiles use the
// guarded synchronous path.
// ======================================================================
template <bool WT>
__global__ __launch_bounds__(256) void gemm_wmma_f32(
    const float* __restrict__ A, const float* __restrict__ W,
    const float* __restrict__ bias1, const float* __restrict__ bias2,
    float* __restrict__ C, int M, int N, int K, int ldc,
    int relu, int splitK, int atomicOut) {
  __shared__ float As[2][128][17];   // +1 pad: conflict-free fragment reads
  __shared__ float Ws[2][16][65];

  const int tid = threadIdx.x;
  const int lane = tid & 31;
  const int wid = tid >> 5;
  const int half = lane >> 4;     // which 16-lane half of the wave
  const int lm = lane & 15;
  const int waveM = (wid >> 1) * 32;  // 0,32,64,96
  const int waveN = (wid & 1) * 32;   // 0,32
  const int blockM = blockIdx.y * 128;
  const int blockN = blockIdx.x * 64;

  long long chunk = ((long long)K + splitK - 1) / splitK;
  chunk = (chunk + 15) & ~15LL;
  long long kBeg = (long long)blockIdx.z * chunk;
  long long kEnd = kBeg + chunk;
  if (kEnd > K) kEnd = K;

  v8f acc[2][2] = {};

  // ---- fragment compute on one staged 128x16 (A) / 16x64 (W) tile ----
  auto compute_tile = [&](const float (*At)[17], const float (*Wt)[65]) {
#pragma unroll
    for (int kk = 0; kk < 16; kk += 4) {
#if HAVE_WMMA_F32
      // A frag (16x4): VGPR0 = K0 (lanes 0-15) / K2 (lanes 16-31); VGPR1 = K1/K3
      v2f a[2], b[2];
#pragma unroll
      for (int r = 0; r < 2; ++r) {
        int mrow = waveM + r * 16 + lm;
        a[r].x = At[mrow][kk + 2 * half];
        a[r].y = At[mrow][kk + 2 * half + 1];
      }
#pragma unroll
      for (int c = 0; c < 2; ++c) {
        int ncol = waveN + c * 16 + lm;
        b[c].x = Wt[kk + 2 * half][ncol];
        b[c].y = Wt[kk + 2 * half + 1][ncol];
      }
#pragma unroll
      for (int r = 0; r < 2; ++r)
#pragma unroll
        for (int c = 0; c < 2; ++c)
          acc[r][c] = __builtin_amdgcn_wmma_f32_16x16x4_f32(
              false, a[r], false, b[c], (short)0, acc[r][c], false, false);
#else
      // scalar fallback with identical C-fragment layout
#pragma unroll
      for (int r = 0; r < 2; ++r)
#pragma unroll
        for (int c = 0; c < 2; ++c)
#pragma unroll
          for (int v = 0; v < 8; ++v) {
            int row = waveM + r * 16 + v + (half ? 8 : 0);
            int col = waveN + c * 16 + lm;
            float s = acc[r][c][v];
#pragma unroll
            for (int k4 = 0; k4 < 4; ++k4) s += At[row][kk + k4] * Wt[kk + k4][col];
            acc[r][c][v] = s;
          }
#endif
    }
  };

#if HAVE_WMMA_F32 && HAVE_ASYNC_LDS
  // Async double-buffered pipeline for fully aligned tiles.
  // (M-partial blocks read extra rows that stay inside the workspace and are
  //  never stored; their wave tiles are entirely outside the store mask.)
  const bool fullTile =
      (kEnd > kBeg) && (((kEnd - kBeg) & 15) == 0) && (blockN + 64 <= N) && (blockM < M);
  if (fullTile) {
    const int nst = (int)((kEnd - kBeg) >> 4);
    auto issue = [&](int stage) {
      const int buf = stage & 1;
      const long long k0 = kBeg + ((long long)stage << 4);
#pragma unroll
      for (int i = 0; i < 8; ++i) {           // A tile: 8 b32 async loads/thread
        const int idx = tid + i * 256;
        const int r = idx >> 4, c = idx & 15;
        async_ld32(&A[(long long)(blockM + r) * K + k0 + c], &As[buf][r][c]);
      }
#pragma unroll
      for (int i = 0; i < 4; ++i) {           // W tile: 4 b32 async loads/thread
        const int idx = tid + i * 256;
        const int r = idx >> 6, c = idx & 63;
        const float* gp = WT ? &W[(long long)(blockN + c) * K + k0 + r]
                             : &W[(k0 + r) * (long long)N + blockN + c];
        async_ld32(gp, &Ws[buf][r][c]);
      }
    };
    issue(0);
    for (int s = 0; s < nst; ++s) {
      if (s + 1 < nst) {
        issue(s + 1);
        // 12 next-stage loads in flight; previous stage's 12 complete (in-order)
        asm volatile("s_wait_asynccnt 0xc" ::: "memory");
      } else {
        asm volatile("s_wait_asynccnt 0x0" ::: "memory");
      }
      __syncthreads();                 // staged data visible to all waves
      compute_tile(As[s & 1], Ws[s & 1]);
      __syncthreads();                 // all reads done before buffer reuse
    }
  } else
#endif
  {
    for (long long k0 = kBeg; k0 < kEnd; k0 += 16) {
      // ---- stage A tile 128x16 ----
#pragma unroll
      for (int i = 0; i < 8; ++i) {
        int idx = tid + i * 256;
        int r = idx >> 4, c = idx & 15;
        long long k = k0 + c;
        int m = blockM + r;
        As[0][r][c] = (m < M && k < kEnd) ? A[(long long)m * K + k] : 0.0f;
      }
      // ---- stage W tile 16x64 ----
#pragma unroll
      for (int i = 0; i < 4; ++i) {
        int idx = tid + i * 256;
        int r = idx >> 6, c = idx & 63;
        long long k = k0 + r;
        int n = blockN + c;
        float v = 0.0f;
        if (k < kEnd && n < N)
          v = WT ? W[(long long)n * K + k] : W[k * (long long)N + n];
        Ws[0][r][c] = v;
      }
      // prefetch next A tile (gfx1250 global_prefetch_b8 path)
      if (k0 + 16 < kEnd) {
        int r = tid >> 1;
        int m = blockM + r;
        if (m < M) __builtin_prefetch(&A[(long long)m * K + k0 + 16], 0, 1);
      }
      __syncthreads();
      compute_tile(As[0], Ws[0]);
      __syncthreads();
    }
  }

  // ---- store: C/D layout: VGPR v -> M = v + 8*half, N = lane%16 ----
#pragma unroll
  for (int r = 0; r < 2; ++r)
#pragma unroll
    for (int c = 0; c < 2; ++c)
#pragma unroll
      for (int v = 0; v < 8; ++v) {
        int row = blockM + waveM + r * 16 + v + (half ? 8 : 0);
        int col = blockN + waveN + c * 16 + lm;
        if (row < M && col < N) {
          float val = acc[r][c][v];
          if (atomicOut) {
            atomicAdd(&C[(long long)row * ldc + col], val);
          } else {
            if (bias1) val += bias1[col];
            if (bias2) val += bias2[col];
            if (relu) val = val > 0.0f ? val : 0.0f;
            C[(long long)row * ldc + col] = val;
          }
        }
      }
}

// ======================================================================
// elementwise / scatter kernels
// ======================================================================
__global__ void k_fill_u32(unsigned* p, long long n, unsigned v) {
  long long i = blockIdx.x * 256LL + threadIdx.x;
  if (i < n) p[i] = v;
}

__global__ void k_bias_act(float* X, const float* __restrict__ b, int cols,
                           long long n, int relu) {
  long long i = blockIdx.x * 256LL + threadIdx.x;
  if (i >= n) return;
  float v = X[i] + b[(int)(i % cols)];
  if (relu) v = v > 0.0f ? v : 0.0f;
  X[i] = v;
}

__global__ void k_residual_relu(float* X, const float* __restrict__ Y, long long n) {
  long long i = blockIdx.x * 256LL + threadIdx.x;
  if (i >= n) return;
  float v = X[i] + Y[i];
  X[i] = v > 0.0f ? v : 0.0f;
}

// asrc[n,h] / adst[n,h] head dots
__global__ void k_headdots(const float* __restrict__ h, const float* __restrict__ a_s,
                           const float* __restrict__ a_d, float* asrc, float* adst) {
  int idx = blockIdx.x * 256 + threadIdx.x;
  if (idx >= Nn * HEADS) return;
  int n = idx / HEADS, hd = idx % HEADS;
  const float* hp = h + (long long)n * F780 + hd * F0;
  const float* ws = a_s + hd * F0;
  const float* wd = a_d + hd * F0;
  float s = 0.f, d = 0.f;
#pragma unroll 6
  for (int f = 0; f < F0; ++f) { float hv = hp[f]; s += hv * ws[f]; d += hv * wd[f]; }
  asrc[idx] = s;
  adst[idx] = d;
}

__global__ void k_edge_logit_max(const int* __restrict__ ei, const float* __restrict__ asrc,
                                 const float* __restrict__ adst, float* ework,
                                 unsigned* mseg) {
  int idx = blockIdx.x * 256 + threadIdx.x;
  if (idx >= E2c * HEADS) return;
  int e2 = idx / HEADS, hd = idx % HEADS;
  int s, d; edge_sd(ei, e2, s, d);
  float e = asrc[s * HEADS + hd] + adst[d * HEADS + hd];
  e = e > 0.f ? e : 0.2f * e;                // leaky_relu(0.2)
  ework[idx] = e;
  atomicMax(&mseg[d * HEADS + hd], fordu(e));
}

__global__ void k_edge_exp_sum(const int* __restrict__ ei, const unsigned* __restrict__ mseg,
                               float* ework, float* denom) {
  int idx = blockIdx.x * 256 + threadIdx.x;
  if (idx >= E2c * HEADS) return;
  int e2 = idx / HEADS, hd = idx % HEADS;
  int s, d; edge_sd(ei, e2, s, d);
  float p = expf(ework[idx] - unford(mseg[d * HEADS + hd]));
  ework[idx] = p;
  atomicAdd(&denom[d * HEADS + hd], p);
}

__global__ void k_gat_aggregate(const int* __restrict__ ei, const float* __restrict__ ework,
                                const float* __restrict__ denom, const float* __restrict__ h,
                                float* xg) {
  long long idx = blockIdx.x * 256LL + threadIdx.x;
  if (idx >= (long long)E2c * F780) return;
  int e2 = (int)(idx / F780), r = (int)(idx % F780);
  int hd = r / F0;
  int s, d; edge_sd(ei, e2, s, d);
  float alpha = ework[e2 * HEADS + hd] / denom[d * HEADS + hd];
  atomicAdd(&xg[(long long)d * F780 + r], alpha * h[(long long)s * F780 + r]);
}

__global__ void k_deg(const int* __restrict__ ei, float* deg) {
  int e2 = blockIdx.x * 256 + threadIdx.x;
  if (e2 >= E2c) return;
  int s, d; edge_sd(ei, e2, s, d);
  atomicAdd(&deg[d], 1.0f);
}
__global__ void k_rsqrt(const float* __restrict__ deg, float* dn) {
  int i = blockIdx.x * 256 + threadIdx.x;
  if (i < Nn) dn[i] = rsqrtf(deg[i]);
}

__global__ void k_scatter_norm(const int* __restrict__ ei, const float* __restrict__ dn,
                               const float* __restrict__ hz, float* out, int F) {
  long long idx = blockIdx.x * 256LL + threadIdx.x;
  if (idx >= (long long)E2c * F) return;
  int e2 = (int)(idx / F), f = (int)(idx % F);
  int s, d; edge_sd(ei, e2, s, d);
  atomicAdd(&out[(long long)d * F + f], hz[(long long)s * F + f] * dn[s] * dn[d]);
}

__global__ void k_pool_max(const float* __restrict__ x, const int* __restrict__ batch,
                           unsigned* g) {
  long long idx = blockIdx.x * 256LL + threadIdx.x;
  if (idx >= (long long)Nn * F3120) return;
  int n = (int)(idx / F3120), f = (int)(idx % F3120);
  atomicMax(&g[(long long)batch[n] * F3120 + f], fordu(x[idx]));
}
__global__ void k_decode(const unsigned* __restrict__ g, float* out, long long n) {
  long long i = blockIdx.x * 256LL + threadIdx.x;
  if (i < n) out[i] = unford(g[i]);
}

__global__ void k_embed(const int* __restrict__ target, const float* __restrict__ emb,
                        float* et) {
  int idx = blockIdx.x * 256 + threadIdx.x;
  if (idx >= Bb * Ts * 128) return;
  int c = idx & 127, bt = idx >> 7;
  et[idx] = emb[target[bt] * 128 + c];
}

// ======================================================================
// Persistent LSTM scan: one workgroup per direction; h-state in LDS,
// c-state in registers; gx (input gates) precomputed by WMMA GEMMs.
// PyTorch gate row order: i, f, g, o  (rows 0..511 of whh).
// ======================================================================
__global__ __launch_bounds__(512) void lstm_scan(
    const float* __restrict__ gxF, const float* __restrict__ gxB,
    const float* __restrict__ whh,   // [2,512,128]
    const float* __restrict__ h0,    // [2,32,128] for this layer
    const float* __restrict__ c0, float* __restrict__ y) {
  const int dir = blockIdx.x;
  const float* gx = dir ? gxB : gxF;
  const float* wh = whh + (long long)dir * 512 * 128;
  __shared__ float hs[Bb * 128];
  const int tid = threadIdx.x;

  for (int i = tid; i < Bb * 128; i += 512) hs[i] = h0[dir * Bb * 128 + i];
  float creg[8];
#pragma unroll
  for (int k = 0; k < 8; ++k) creg[k] = c0[dir * Bb * 128 + tid + k * 512];
  __syncthreads();

  for (int s = 0; s < Ts; ++s) {
    const int t = dir ? (Ts - 1 - s) : s;
    float hn[8];
#pragma unroll
    for (int k = 0; k < 8; ++k) {
      const int idx = tid + k * 512;       // (b,u)
      const int b = idx >> 7, u = idx & 127;
      const float* hb = &hs[b * 128];
      const float* g0 = &gx[((long long)b * Ts + t) * 512];
      float ai = g0[u], af = g0[128 + u], ag = g0[256 + u], ao = g0[384 + u];
      const float* wi = &wh[(long long)u * 128];
      const float* wf = &wh[(long long)(128 + u) * 128];
      const float* wg = &wh[(long long)(256 + u) * 128];
      const float* wo = &wh[(long long)(384 + u) * 128];
#pragma unroll 4
      for (int kk = 0; kk < 128; ++kk) {
        float hv = hb[kk];
        ai += hv * wi[kk]; af += hv * wf[kk];
        ag += hv * wg[kk]; ao += hv * wo[kk];
      }
      float cn = sigf(af) * creg[k] + sigf(ai) * tanhf(ag);
      creg[k] = cn;
      hn[k] = sigf(ao) * tanhf(cn);
    }
    __syncthreads();
#pragma unroll
    for (int k = 0; k < 8; ++k) {
      const int idx = tid + k * 512;
      const int b = idx >> 7, u = idx & 127;
      hs[idx] = hn[k];
      y[((long long)b * Ts + t) * 256 + dir * 128 + u] = hn[k];
    }
    __syncthreads();
  }
}

__global__ void k_final(const float* __restrict__ xc2, const float* __restrict__ w,
                        const float* __restrict__ b, float* out) {
  int bi = threadIdx.x;
  if (bi >= Bb) return;
  float s = b[0];
  for (int j = 0; j < 512; ++j) s += xc2[bi * 512 + j] * w[j];
  out[bi] = s;
}

// ======================================================================
// host side
// ======================================================================
static inline void launch_gemm(hipStream_t s, const float* A, const float* W,
                               const float* b1, const float* b2, float* C, int M,
                               int N, int K, int ldc, int relu, bool wt,
                               int splitK, int atomicOut) {
  dim3 grid((N + 63) / 64, (M + 127) / 128, splitK);
  if (wt)
    gemm_wmma_f32<true><<<grid, 256, 0, s>>>(A, W, b1, b2, C, M, N, K, ldc, relu, splitK, atomicOut);
  else
    gemm_wmma_f32<false><<<grid, 256, 0, s>>>(A, W, b1, b2, C, M, N, K, ldc, relu, splitK, atomicOut);
}
static inline dim3 ew_grid(long long n) { return dim3((unsigned)((n + 255) / 256)); }

extern "C" void kernel_launch(void* const* d_in, const int* in_sizes, int n_in,
                              void* d_out, int out_size, void* d_ws, size_t ws_size,
                              hipStream_t stream) {
  (void)in_sizes; (void)n_in; (void)out_size; (void)ws_size;
  const float* x        = (const float*)d_in[0];
  const int*   ei       = (const int*)d_in[1];
  const int*   batch    = (const int*)d_in[2];
  const int*   target   = (const int*)d_in[3];
  const float* hidden   = (const float*)d_in[4];
  const float* cell     = (const float*)d_in[5];
  const float* gat_w    = (const float*)d_in[6];
  const float* gat_as   = (const float*)d_in[7];
  const float* gat_ad   = (const float*)d_in[8];
  const float* gat_b    = (const float*)d_in[9];
  const float* gcn2_w   = (const float*)d_in[10];
  const float* gcn2_b   = (const float*)d_in[11];
  const float* gcn3_w   = (const float*)d_in[12];
  const float* gcn3_b   = (const float*)d_in[13];
  const float* rb_w     = (const float*)d_in[14];
  const float* rb_b     = (const float*)d_in[15];
  const float* rb_ln    = (const float*)d_in[16];
  const float* fcg1_w   = (const float*)d_in[17];
  const float* fcg1_b   = (const float*)d_in[18];
  const float* fcg2_w   = (const float*)d_in[19];
  const float* fcg2_b   = (const float*)d_in[20];
  const float* emb      = (const float*)d_in[21];
  const float* l0_wih   = (const float*)d_in[22];
  const float* l0_whh   = (const float*)d_in[23];
  const float* l0_bih   = (const float*)d_in[24];
  const float* l0_bhh   = (const float*)d_in[25];
  const float* l1_wih   = (const float*)d_in[26];
  const float* l1_whh   = (const float*)d_in[27];
  const float* l1_bih   = (const float*)d_in[28];
  const float* l1_bhh   = (const float*)d_in[29];
  const float* fxt1_w   = (const float*)d_in[30];
  const float* fxt1_b   = (const float*)d_in[31];
  const float* fxt2_w   = (const float*)d_in[32];
  const float* fxt2_b   = (const float*)d_in[33];
  const float* fc1_w    = (const float*)d_in[34];
  const float* fc1_b    = (const float*)d_in[35];
  const float* fc2_w    = (const float*)d_in[36];
  const float* fc2_b    = (const float*)d_in[37];
  const float* out_w    = (const float*)d_in[38];
  const float* out_b    = (const float*)d_in[39];
  float* out = (float*)d_out;
  char*  ws  = (char*)d_ws;

  size_t off = 0;
  auto alloc = [&](size_t elems) -> float* {
    float* p = (float*)(ws + off);
    off += ((elems * sizeof(float)) + 255) & ~size_t(255);
    return p;
  };

  // ---- persistent small buffers ----
  float* xcat = alloc(32 * 256);
  float* xc1  = alloc(32 * 1024);
  float* xc2  = alloc(32 * 512);
  float* xt1  = alloc(32 * 1500);
  const size_t arenaBase = off;

  // ---- graph-phase arena ----
  float* hbuf   = alloc((size_t)Nn * F780);
  float* xg     = alloc((size_t)Nn * F780);
  float* asrc   = alloc((size_t)Nn * HEADS);
  float* adst   = alloc((size_t)Nn * HEADS);
  float* msegF  = alloc((size_t)Nn * HEADS);   unsigned* mseg = (unsigned*)msegF;
  float* denom  = alloc((size_t)Nn * HEADS);
  float* deg    = alloc(Nn);
  float* dn     = alloc(Nn);
  float* ework  = alloc((size_t)E2c * HEADS);
  float* hz1    = alloc((size_t)Nn * F1560);
  float* t1     = alloc((size_t)Nn * F1560);
  float* big3a  = alloc((size_t)Nn * F3120);
  float* big3b  = alloc((size_t)Nn * F3120);
  float* big3c  = alloc((size_t)Nn * F3120);
  float* gpoolF = alloc((size_t)Bb * F3120);   unsigned* gpool = (unsigned*)gpoolF;
  float* gdec   = alloc((size_t)Bb * F3120);
  float* fcg1o  = alloc((size_t)Bb * 1500);

  const unsigned NEG_INF_ORD = 0x007FFFFFu;  // fordu(-inf)

  // ================= graph branch =================
  // h = x @ gat_w  [N,780]
  launch_gemm(stream, x, gat_w, nullptr, nullptr, hbuf, Nn, F780, F0, F780, 0, false, 1, 0);
  k_headdots<<<ew_grid((long long)Nn * HEADS), 256, 0, stream>>>(hbuf, gat_as, gat_ad, asrc, adst);
  k_fill_u32<<<ew_grid((long long)Nn * HEADS), 256, 0, stream>>>(mseg, (long long)Nn * HEADS, NEG_INF_ORD);
  k_edge_logit_max<<<ew_grid((long long)E2c * HEADS), 256, 0, stream>>>(ei, asrc, adst, ework, mseg);
  (void)hipMemsetAsync(denom, 0, (size_t)Nn * HEADS * 4, stream);
  k_edge_exp_sum<<<ew_grid((long long)E2c * HEADS), 256, 0, stream>>>(ei, mseg, ework, denom);
  (void)hipMemsetAsync(xg, 0, (size_t)Nn * F780 * 4, stream);
  k_gat_aggregate<<<ew_grid((long long)E2c * F780), 256, 0, stream>>>(ei, ework, denom, hbuf, xg);
  k_bias_act<<<ew_grid((long long)Nn * F780), 256, 0, stream>>>(xg, gat_b, F780, (long long)Nn * F780, 1);

  // GCN degree normalization (shared)
  (void)hipMemsetAsync(deg, 0, (size_t)Nn * 4, stream);
  k_deg<<<ew_grid(E2c), 256, 0, stream>>>(ei, deg);
  k_rsqrt<<<ew_grid(Nn), 256, 0, stream>>>(deg, dn);

  // gcn2: relu(scatter(xg@W)*norm + b)
  launch_gemm(stream, xg, gcn2_w, nullptr, nullptr, hz1, Nn, F1560, F780, F1560, 0, false, 1, 0);
  (void)hipMemsetAsync(t1, 0, (size_t)Nn * F1560 * 4, stream);
  k_scatter_norm<<<ew_grid((long long)E2c * F1560), 256, 0, stream>>>(ei, dn, hz1, t1, F1560);
  k_bias_act<<<ew_grid((long long)Nn * F1560), 256, 0, stream>>>(t1, gcn2_b, F1560, (long long)Nn * F1560, 1);

  // gcn3
  launch_gemm(stream, t1, gcn3_w, nullptr, nullptr, big3c, Nn, F3120, F1560, F3120, 0, false, 1, 0);
  (void)hipMemsetAsync(big3a, 0, (size_t)Nn * F3120 * 4, stream);
  k_scatter_norm<<<ew_grid((long long)E2c * F3120), 256, 0, stream>>>(ei, dn, big3c, big3a, F3120);
  k_bias_act<<<ew_grid((long long)Nn * F3120), 256, 0, stream>>>(big3a, gcn3_b, F3120, (long long)Nn * F3120, 1);

  // 4x residual block (shared weights):
  // xg = relu( relu(gcn(xg,rb_w,rb_b)) @ rb_ln + xg )
  for (int it = 0; it < 4; ++it) {
    launch_gemm(stream, big3a, rb_w, nullptr, nullptr, big3c, Nn, F3120, F3120, F3120, 0, false, 1, 0);
    (void)hipMemsetAsync(big3b, 0, (size_t)Nn * F3120 * 4, stream);
    k_scatter_norm<<<ew_grid((long long)E2c * F3120), 256, 0, stream>>>(ei, dn, big3c, big3b, F3120);
    k_bias_act<<<ew_grid((long long)Nn * F3120), 256, 0, stream>>>(big3b, rb_b, F3120, (long long)Nn * F3120, 1);
    launch_gemm(stream, big3b, rb_ln, nullptr, nullptr, big3c, Nn, F3120, F3120, F3120, 0, false, 1, 0);
    k_residual_relu<<<ew_grid((long long)Nn * F3120), 256, 0, stream>>>(big3a, big3c, (long long)Nn * F3120);
  }

  // global max pool + graph MLP head -> xcat[:, 0:128]
  k_fill_u32<<<ew_grid((long long)Bb * F3120), 256, 0, stream>>>(gpool, (long long)Bb * F3120, NEG_INF_ORD);
  k_pool_max<<<ew_grid((long long)Nn * F3120), 256, 0, stream>>>(big3a, batch, gpool);
  k_decode<<<ew_grid((long long)Bb * F3120), 256, 0, stream>>>(gpool, gdec, (long long)Bb * F3120);
  launch_gemm(stream, gdec, fcg1_w, fcg1_b, nullptr, fcg1o, Bb, 1500, F3120, 1500, 1, false, 1, 0);
  launch_gemm(stream, fcg1o, fcg2_w, fcg2_b, nullptr, xcat, Bb, 128, 1500, 256, 0, false, 1, 0);

  // ================= protein branch (reuse arena) =================
  off = arenaBase;
  float* et  = alloc((size_t)Bb * Ts * 128);
  float* y0  = alloc((size_t)Bb * Ts * 256);
  float* y1  = alloc((size_t)Bb * Ts * 256);
  float* gxA = alloc((size_t)Bb * Ts * 512);
  float* gxB = alloc((size_t)Bb * Ts * 512);

  k_embed<<<ew_grid((long long)Bb * Ts * 128), 256, 0, stream>>>(target, emb, et);

  // layer 0: gx = et @ wih^T + bih + bhh  (wih stored [512,128] -> WT)
  launch_gemm(stream, et, l0_wih,             l0_bih,       l0_bhh,       gxA, Bb * Ts, 512, 128, 512, 0, true, 1, 0);
  launch_gemm(stream, et, l0_wih + 512 * 128, l0_bih + 512, l0_bhh + 512, gxB, Bb * Ts, 512, 128, 512, 0, true, 1, 0);
  lstm_scan<<<2, 512, 0, stream>>>(gxA, gxB, l0_whh, hidden, cell, y0);

  // layer 1 (input 256 = concat fwd/bwd)
  launch_gemm(stream, y0, l1_wih,             l1_bih,       l1_bhh,       gxA, Bb * Ts, 512, 256, 512, 0, true, 1, 0);
  launch_gemm(stream, y0, l1_wih + 512 * 256, l1_bih + 512, l1_bhh + 512, gxB, Bb * Ts, 512, 256, 512, 0, true, 1, 0);
  lstm_scan<<<2, 512, 0, stream>>>(gxA, gxB, l1_whh, hidden + 2 * Bb * 128, cell + 2 * Bb * 128, y1);

  // fxt1: [32,256000] @ [256000,1500]  -- HBM bound (1.54 GB weights),
  // split-K=80 spreads the weight stream across 1920 workgroups.
  (void)hipMemsetAsync(xt1, 0, (size_t)Bb * 1500 * 4, stream);
  launch_gemm(stream, y1, fxt1_w, nullptr, nullptr, xt1, Bb, 1500, Ts * 256, 1500, 0, false, 80, 1);
  k_bias_act<<<ew_grid((long long)Bb * 1500), 256, 0, stream>>>(xt1, fxt1_b, 1500, (long long)Bb * 1500, 0);
  launch_gemm(stream, xt1, fxt2_w, fxt2_b, nullptr, xcat + 128, Bb, 128, 1500, 256, 0, false, 1, 0);

  // ================= fusion head =================
  launch_gemm(stream, xcat, fc1_w, fc1_b, nullptr, xc1, Bb, 1024, 256, 1024, 1, false, 1, 0);
  launch_gemm(stream, xc1,  fc2_w, fc2_b, nullptr, xc2, Bb, 512, 1024, 512, 1, false, 1, 0);
  k_final<<<1, 32, 0, stream>>>(xc2, out_w, out_b, out);
}